// ProposalTarget_1649267441863
// MI455X (gfx1250) — compile-verified
//
#include <hip/hip_runtime.h>

// RPN proposal decode for MI455X (gfx1250).
// Bandwidth-bound elementwise kernel (~145 MB traffic @ 23.3 TB/s ≈ 6.3 us).
// CDNA5-specific paths: async global->LDS (ASYNCcnt) anchor staging +
// s_wait_asynccnt; LDS-staged, b128-vectorized, non-temporal coalesced
// output stores. No matrix structure exists in this op, so WMMA is
// intentionally not used.

typedef float v4f __attribute__((ext_vector_type(4)));   // native vector for NT builtins

namespace {
constexpr int   kB = 16;
constexpr int   kA = 9;
constexpr int   kH = 128;
constexpr int   kW = 128;
constexpr int   kHW = kH * kW;
constexpr float kS = 2048.0f;
constexpr float kInvS = 1.0f / 2048.0f;   // exact (power of two)
constexpr float kThresh = 0.7f;
}

__global__ __launch_bounds__(256)
void proposal_decode_kernel(const float* __restrict__ cla,
                            const float* __restrict__ reg,
                            const float* __restrict__ anchor,
                            float* __restrict__ out)
{
    // acx, acy, aw, ah for each w of this block's 'a'  (2 KB)
    __shared__ __align__(16) float s_anc[kW * 4];
    // 9 outputs per thread, staged for coalesced stores (9 KB)
    __shared__ __align__(16) float s_out[256 * 9];

    const int tid = threadIdx.x;
    const int bi  = blockIdx.x;

    const int h2 = bi & 63;          // H/2 = 64 blocks per (b,a)
    const int ba = bi >> 6;
    const int a  = ba % kA;
    const int b  = ba / kA;

    const int hh = tid >> 7;         // 0 or 1
    const int w  = tid & (kW - 1);
    const int h  = 2 * h2 + hh;

    // ---- Stage anchor params via async global->LDS (gfx1250 path) --------
    // Reference gather: idx[a][w] = w*(W+1)*a ; we need anchor[idx][2..5].
    // Lanes with tid<128 fetch elems {2,3} (acx,acy); tid>=128 fetch {4,5}
    // (aw,ah). Both global and LDS addresses are 8-byte aligned.
    {
        const int ww   = tid & (kW - 1);
        const int half = tid >> 7;                      // 0 or 1
        const int row  = ww * (kW + 1) * a;
        const unsigned long long gaddr =
            (unsigned long long)(anchor + row * 6 + 2 + half * 2);
        // Low 32 bits of a generic LDS pointer are the LDS byte offset.
        const unsigned lds_off =
            (unsigned)(unsigned long long)(const void*)&s_anc[ww * 4 + half * 2];
        asm volatile("global_load_async_to_lds_b64 %0, %1, off"
                     :: "v"(lds_off), "v"(gaddr)
                     : "memory");
    }
    __builtin_amdgcn_s_wait_asynccnt(0);
    __syncthreads();

    // ---- Classification: fg = softmax(ch)[1] = sigmoid(c1 - c0) ----------
    // Streaming single-pass data: non-temporal loads (don't churn L2).
    const int cbase = (b * (2 * kA) + 2 * a) * kHW + h * kW + w;
    const float c0 = __builtin_nontemporal_load(cla + cbase);
    const float c1 = __builtin_nontemporal_load(cla + cbase + kHW);
    const float fg = 1.0f / (1.0f + __expf(c0 - c1));

    // ---- Regression channels (all coalesced b32 loads) -------------------
    const int rbase = (b * (4 * kA) + 4 * a) * kHW + h * kW + w;
    const float tx = __builtin_nontemporal_load(reg + rbase);
    const float ty = __builtin_nontemporal_load(reg + rbase + kHW);
    const float tw = __builtin_nontemporal_load(reg + rbase + 2 * kHW);
    const float th = __builtin_nontemporal_load(reg + rbase + 3 * kHW);

    const float acx = s_anc[w * 4 + 0];
    const float acy = s_anc[w * 4 + 1];
    const float aw  = s_anc[w * 4 + 2];
    const float ah  = s_anc[w * 4 + 3];

    // ---- Box decode ------------------------------------------------------
    const float cx = (tx * aw + acx) * kS;
    const float cy = (ty * ah + acy) * kS;
    const float bw = __expf(tw) * aw * kS;
    const float bh = __expf(th) * ah * kS;

    const float ltx = cx - bw * 0.5f;
    const float lty = cy - bh * 0.5f;
    const float rbx = cx + bw * 0.5f;
    const float rby = cy + bh * 0.5f;

    const bool valid = (fg > kThresh) && (ltx >= 0.0f) && (lty >= 0.0f) &&
                       (rbx <= kS) && (rby <= kS);
    const float m = valid ? 1.0f : 0.0f;

    // ---- Stage 9 channels into LDS (stride 9 is conflict-free mod 64) ----
    const int sb = tid * 9;
    s_out[sb + 0] = ltx * m;
    s_out[sb + 1] = lty * m;
    s_out[sb + 2] = rbx * m;
    s_out[sb + 3] = rby * m;
    s_out[sb + 4] = (cx * kInvS) * m;
    s_out[sb + 5] = (cy * kInvS) * m;
    s_out[sb + 6] = (bw * kInvS) * m;
    s_out[sb + 7] = (bh * kInvS) * m;
    s_out[sb + 8] = m;
    __syncthreads();

    // ---- Coalesced b128 store of the block's 2304 contiguous floats ------
    // 576 x 16B chunks per block; ds_load_b128 + global_store_b128, both
    // fully coalesced. Output is write-once streaming -> non-temporal.
    const v4f* s4 = reinterpret_cast<const v4f*>(s_out);
    v4f* out4 = reinterpret_cast<v4f*>(out + bi * (256 * 9));
#pragma unroll
    for (int i = 0; i < 3; ++i) {
        const int idx = i * 256 + tid;
        if (idx < 576) {
            __builtin_nontemporal_store(s4[idx], out4 + idx);
        }
    }
}

extern "C" void kernel_launch(void* const* d_in, const int* in_sizes, int n_in,
                              void* d_out, int out_size, void* d_ws, size_t ws_size,
                              hipStream_t stream)
{
    (void)in_sizes; (void)n_in; (void)out_size; (void)d_ws; (void)ws_size;

    const float* cla    = (const float*)d_in[0];   // (16, 18, 128, 128) f32
    const float* reg    = (const float*)d_in[1];   // (16, 36, 128, 128) f32
    const float* anchor = (const float*)d_in[2];   // (128*128*9, 6) f32
    // d_in[3] = src_info (2048), d_in[4] = num_anchors (9): fixed by setup.
    float* out = (float*)d_out;                    // (16, 9, 128, 128, 9) f32

    const int grid = kB * kA * (kH / 2);           // 9216 blocks
    proposal_decode_kernel<<<grid, 256, 0, stream>>>(cla, reg, anchor, out);
}